// LipophilicityGNN_730144440677
// MI455X (gfx1250) — compile-verified
//
#include <hip/hip_runtime.h>
#include <hip/hip_bf16.h>

typedef float v2f __attribute__((ext_vector_type(2)));
typedef float v8f __attribute__((ext_vector_type(8)));

#define H_CH 64

// ---------------------------------------------------------------------------
// Degree / normalization
// ---------------------------------------------------------------------------
__global__ __launch_bounds__(256) void deg_kernel(const int* __restrict__ dst,
                                                  float* __restrict__ deg, int E) {
    int e = blockIdx.x * 256 + threadIdx.x;
    if (e < E) atomicAdd(&deg[dst[e]], 1.0f);
}

__global__ __launch_bounds__(256) void dinv_kernel(const float* __restrict__ deg,
                                                   float* __restrict__ dinv, int N) {
    int i = blockIdx.x * 256 + threadIdx.x;
    if (i < N) dinv[i] = rsqrtf(deg[i] + 1.0f);
}

// ---------------------------------------------------------------------------
// GEMM  Y[M,64] = X[M,K] @ W[K,64]   (fp32, native f32 WMMA, 16x16x4)
// One wave -> one 16x64 output tile (4 x v8f accumulators).
//
// A layout (16x4 f32): lane l holds row M=l&15; VGPR0/1 = K = kbase, kbase+1
//   where kbase = (l>=16) ? k+2 : k.   -> contiguous float2 from X row.
// B layout (4x16 f32): lane l holds column N=l&15; VGPR0/1 = same K rows.
//   W is staged in LDS K-pair interleaved: sW[(k/2)*128 + col*2 + (k&1)],
//   so each B fragment is ONE aligned ds_load_b64 directly into an
//   even-aligned VGPR pair (WMMA operand requirement), bank-conflict free.
// C layout (16x16 f32): VGPR r, lane l -> M = r + (l>>4)*8, N = l&15.
// ---------------------------------------------------------------------------
template <int K>
__global__ __launch_bounds__(256) void gemm_wmma_kernel(const float* __restrict__ X,
                                                        const float* __restrict__ W,
                                                        float* __restrict__ Y, int M) {
    __shared__ v2f sW[(K / 2) * H_CH];   // [K/2][64] of (k_even, k_odd) pairs
    {
        float* sWf = (float*)sW;
        for (int i = threadIdx.x; i < K * H_CH; i += 256) {
            const int k = i >> 6;       // K row
            const int c = i & 63;       // column
            sWf[(k >> 1) * (2 * H_CH) + c * 2 + (k & 1)] = W[i];
        }
    }
    __syncthreads();

    const int lane = threadIdx.x & 31;
    const int wave = threadIdx.x >> 5;
    const int tile = blockIdx.x * 8 + wave;
    const int tiles = (M + 15) >> 4;
    if (tile >= tiles) return;
    const int m0 = tile * 16;

    const int half = lane >> 4;    // 0: K pair (k,k+1), 1: K pair (k+2,k+3)
    const int mrow = lane & 15;    // A row / B column
    int arow = m0 + mrow;
    if (arow >= M) arow = M - 1;   // safe clamp (M % 16 == 0 in practice)
    const float* __restrict__ xrow = X + (size_t)arow * K;

    v8f acc0 = {}, acc1 = {}, acc2 = {}, acc3 = {};

    for (int k = 0; k < K; k += 4) {
        const int kk = k + half * 2;          // even
        v2f a = *reinterpret_cast<const v2f*>(xrow + kk);   // global_load_b64

        const v2f* __restrict__ wp = &sW[(kk >> 1) * H_CH + mrow];
        v2f b0 = wp[0];    // ds_load_b64, lanes stride 8B -> 32 banks per half-wave (halves 512B apart)
        v2f b1 = wp[16];
        v2f b2 = wp[32];
        v2f b3 = wp[48];

        acc0 = __builtin_amdgcn_wmma_f32_16x16x4_f32(false, a, false, b0, (short)0, acc0, false, false);
        acc1 = __builtin_amdgcn_wmma_f32_16x16x4_f32(false, a, false, b1, (short)0, acc1, false, false);
        acc2 = __builtin_amdgcn_wmma_f32_16x16x4_f32(false, a, false, b2, (short)0, acc2, false, false);
        acc3 = __builtin_amdgcn_wmma_f32_16x16x4_f32(false, a, false, b3, (short)0, acc3, false, false);
    }

#pragma unroll
    for (int r = 0; r < 8; ++r) {
        const int m = m0 + half * 8 + r;
        if (m < M) {
            float* __restrict__ yp = Y + (size_t)m * H_CH + mrow;
            yp[0]  = acc0[r];
            yp[16] = acc1[r];
            yp[32] = acc2[r];
            yp[48] = acc3[r];
        }
    }
}

// ---------------------------------------------------------------------------
// Edge aggregation: agg[dst] += h[src] * dinv[src]*dinv[dst]
// 16 threads per edge, float4 gather + 4 atomic f32 adds (resolve in L2).
// ---------------------------------------------------------------------------
__global__ __launch_bounds__(256) void edge_agg_kernel(const int* __restrict__ src,
                                                       const int* __restrict__ dst,
                                                       const float* __restrict__ dinv,
                                                       const float* __restrict__ h,
                                                       float* __restrict__ agg, int E) {
    const int t = blockIdx.x * 256 + threadIdx.x;
    const int e = t >> 4;
    if (e >= E) return;
    const int fo = (t & 15) * 4;
    const int s = src[e];
    const int d = dst[e];
    const float coef = dinv[s] * dinv[d];
    const float4 v = *reinterpret_cast<const float4*>(h + (size_t)s * H_CH + fo);
    float* ap = agg + (size_t)d * H_CH + fo;
    atomicAdd(ap + 0, v.x * coef);
    atomicAdd(ap + 1, v.y * coef);
    atomicAdd(ap + 2, v.z * coef);
    atomicAdd(ap + 3, v.w * coef);
}

// ---------------------------------------------------------------------------
// combine0: h = relu(agg + h0 * dinv^2 + bias)        (out may alias h0)
// ---------------------------------------------------------------------------
__global__ __launch_bounds__(256) void combine0_kernel(const float* __restrict__ agg,
                                                       const float* h0,
                                                       const float* __restrict__ dinv,
                                                       const float* __restrict__ bias,
                                                       float* out, int N) {
    const int t = blockIdx.x * 256 + threadIdx.x;
    if (t >= N * H_CH) return;
    const int i = t >> 6;
    const int f = t & 63;
    const float di = dinv[i];
    const float v = agg[t] + h0[t] * di * di + bias[f];
    out[t] = fmaxf(v, 0.0f);
}

// ---------------------------------------------------------------------------
// combineL: h = relu(agg + hc * dinv^2 + convB) + hs + skipB   (out aliases agg)
// ---------------------------------------------------------------------------
__global__ __launch_bounds__(256) void combineL_kernel(const float* agg,
                                                       const float* __restrict__ hc,
                                                       const float* __restrict__ hs,
                                                       const float* __restrict__ dinv,
                                                       const float* __restrict__ convB,
                                                       const float* __restrict__ skipB,
                                                       float* out, int N) {
    const int t = blockIdx.x * 256 + threadIdx.x;
    if (t >= N * H_CH) return;
    const int i = t >> 6;
    const int f = t & 63;
    const float di = dinv[i];
    const float hn = fmaxf(agg[t] + hc[t] * di * di + convB[f], 0.0f);
    out[t] = hn + hs[t] + skipB[f];
}

// ---------------------------------------------------------------------------
// Global mean pool (sums + counts via atomics)
// ---------------------------------------------------------------------------
__global__ __launch_bounds__(256) void pool_kernel(const float* __restrict__ h,
                                                   const int* __restrict__ b,
                                                   float* __restrict__ sums,
                                                   float* __restrict__ cnts, int N) {
    const int t = blockIdx.x * 256 + threadIdx.x;
    if (t >= N * H_CH) return;
    const int i = t >> 6;
    const int f = t & 63;
    const int g = b[i];
    atomicAdd(&sums[(size_t)g * H_CH + f], h[t]);
    if (f == 0) atomicAdd(&cnts[g], 1.0f);
}

// ---------------------------------------------------------------------------
// Final MLP: out[g] = relu(pooled @ W1 + b1) @ W2 + b2
// ---------------------------------------------------------------------------
__global__ __launch_bounds__(256) void mlp_kernel(const float* __restrict__ sums,
                                                  const float* __restrict__ cnts,
                                                  const float* __restrict__ W1,
                                                  const float* __restrict__ b1,
                                                  const float* __restrict__ W2,
                                                  const float* __restrict__ b2,
                                                  float* __restrict__ out, int G) {
    const int g = blockIdx.x * 256 + threadIdx.x;
    if (g >= G) return;
    const float c = fmaxf(cnts[g], 1.0f);
    float p[H_CH];
#pragma unroll
    for (int k = 0; k < H_CH; ++k) p[k] = sums[(size_t)g * H_CH + k] / c;
    float o = b2[0];
    for (int f = 0; f < 32; ++f) {
        float s = b1[f];
#pragma unroll
        for (int k = 0; k < H_CH; ++k) s += p[k] * W1[k * 32 + f];
        o += fmaxf(s, 0.0f) * W2[f];
    }
    out[g] = o;
}

// ---------------------------------------------------------------------------
// Launch
// ---------------------------------------------------------------------------
extern "C" void kernel_launch(void* const* d_in, const int* in_sizes, int n_in,
                              void* d_out, int out_size, void* d_ws, size_t ws_size,
                              hipStream_t stream) {
    const int F = 128;
    const int N = in_sizes[0] / F;     // 100000 nodes
    const int E = in_sizes[1] / 2;     // 1600000 edges
    const int G = out_size;            // 4096 graphs

    const float* x     = (const float*)d_in[0];
    const int*   eidx  = (const int*)d_in[1];
    const int*   src   = eidx;
    const int*   dst   = eidx + E;
    const int*   b     = (const int*)d_in[2];
    const float* w0    = (const float*)d_in[3];
    const float* b0    = (const float*)d_in[4];
    const float* convW = (const float*)d_in[5];   // [2,64,64]
    const float* convB = (const float*)d_in[6];   // [2,64]
    const float* skipW = (const float*)d_in[7];
    const float* skipB = (const float*)d_in[8];
    const float* lin1W = (const float*)d_in[9];   // [64,32]
    const float* lin1b = (const float*)d_in[10];
    const float* lin2W = (const float*)d_in[11];  // [32,1]
    const float* lin2b = (const float*)d_in[12];
    float* out = (float*)d_out;

    // Workspace layout (floats)
    float* ws   = (float*)d_ws;
    const size_t NH = (size_t)N * H_CH;
    float* bufA = ws;            // node feature buffer
    float* bufB = bufA + NH;
    float* bufC = bufB + NH;
    float* bufD = bufC + NH;
    float* deg  = bufD + NH;     // N
    float* dinv = deg + N;       // N
    float* sums = dinv + N;      // G*64
    float* cnts = sums + (size_t)G * H_CH;  // G

    const int blkN   = (N + 255) / 256;
    const int blkE   = (E + 255) / 256;
    const int blkEF  = (int)(((size_t)E * 16 + 255) / 256);
    const int blkNF  = (int)((NH + 255) / 256);
    const int blkGem = ((N + 15) / 16 + 7) / 8;

    // --- degree / dinv ---
    hipMemsetAsync(deg, 0, (size_t)N * sizeof(float), stream);
    deg_kernel<<<blkE, 256, 0, stream>>>(dst, deg, E);
    dinv_kernel<<<blkN, 256, 0, stream>>>(deg, dinv, N);

    // --- layer 0: h0 = x @ w0 ; agg ; h = relu(agg + h0*dinv^2 + b0) ---
    gemm_wmma_kernel<128><<<blkGem, 256, 0, stream>>>(x, w0, bufA, N);
    hipMemsetAsync(bufB, 0, NH * sizeof(float), stream);
    edge_agg_kernel<<<blkEF, 256, 0, stream>>>(src, dst, dinv, bufA, bufB, E);
    combine0_kernel<<<blkNF, 256, 0, stream>>>(bufB, bufA, dinv, b0, bufA, N);
    // h lives in bufA

    // --- layer 1 ---
    gemm_wmma_kernel<64><<<blkGem, 256, 0, stream>>>(bufA, convW, bufB, N);          // hc
    gemm_wmma_kernel<64><<<blkGem, 256, 0, stream>>>(bufA, skipW, bufC, N);          // hs
    hipMemsetAsync(bufD, 0, NH * sizeof(float), stream);
    edge_agg_kernel<<<blkEF, 256, 0, stream>>>(src, dst, dinv, bufB, bufD, E);
    combineL_kernel<<<blkNF, 256, 0, stream>>>(bufD, bufB, bufC, dinv, convB, skipB, bufD, N);
    // h lives in bufD

    // --- layer 2 ---
    gemm_wmma_kernel<64><<<blkGem, 256, 0, stream>>>(bufD, convW + 64 * 64, bufA, N); // hc
    gemm_wmma_kernel<64><<<blkGem, 256, 0, stream>>>(bufD, skipW + 64 * 64, bufB, N); // hs
    hipMemsetAsync(bufC, 0, NH * sizeof(float), stream);
    edge_agg_kernel<<<blkEF, 256, 0, stream>>>(src, dst, dinv, bufA, bufC, E);
    combineL_kernel<<<blkNF, 256, 0, stream>>>(bufC, bufA, bufB, dinv, convB + 64, skipB + 64, bufC, N);
    // h lives in bufC

    // --- pool + MLP ---
    hipMemsetAsync(sums, 0, (size_t)G * H_CH * sizeof(float), stream);
    hipMemsetAsync(cnts, 0, (size_t)G * sizeof(float), stream);
    pool_kernel<<<blkNF, 256, 0, stream>>>(bufC, b, sums, cnts, N);
    mlp_kernel<<<(G + 255) / 256, 256, 0, stream>>>(sums, cnts, lin1W, lin1b, lin2W, lin2b, out, G);
}